// GCNLayer_2216203125436
// MI455X (gfx1250) — compile-verified
//
#include <hip/hip_runtime.h>

typedef float v2f __attribute__((ext_vector_type(2)));
typedef float v8f __attribute__((ext_vector_type(8)));

#define D 64  // D_IN == D_OUT == 64

// support = X @ W  using V_WMMA_F32_16X16X4_F32.
// One wave per 16x16 output tile; 8 waves (256 thr) per block -> 2 row-tiles x 4 col-tiles.
__global__ void __launch_bounds__(256)
gemm_wmma(const float* __restrict__ X, const float* __restrict__ W,
          float* __restrict__ S, int n)
{
    const int lane = threadIdx.x & 31;
    const int wave = threadIdx.x >> 5;            // 0..7
    const int rowTile = blockIdx.x * 2 + (wave >> 2);
    const int m0 = rowTile * 16;
    if (m0 >= n) return;                          // wave-uniform: EXEC stays all-1s
    const int n0 = (wave & 3) * 16;

    const int h  = lane >> 4;                     // lane half selects K offset (+2)
    const int ml = lane & 15;                     // M (for A) / N (for B,C,D)

    const float* xrow = X + (size_t)(m0 + ml) * D;

    v8f c = {};
    #pragma unroll
    for (int k0 = 0; k0 < D; k0 += 4) {
        const int k = k0 + 2 * h;
        // A tile 16x4 f32: lane ml holds row M=ml; VGPR0=K=k, VGPR1=K=k+1 (contiguous pair)
        v2f a;
        a.x = xrow[k];
        a.y = xrow[k + 1];
        // B tile 4x16 f32: row K striped across lanes; VGPR0=K=k, VGPR1=K=k+1
        v2f b;
        b.x = W[(size_t)k       * D + n0 + ml];
        b.y = W[(size_t)(k + 1) * D + n0 + ml];
        c = __builtin_amdgcn_wmma_f32_16x16x4_f32(
                /*neg_a=*/false, a, /*neg_b=*/false, b,
                /*c_mod=*/(short)0, c, /*reuse_a=*/false, /*reuse_b=*/false);
    }

    // C/D layout: VGPR v -> M = m0 + v + 8*h, N = n0 + ml
    #pragma unroll
    for (int v = 0; v < 8; ++v) {
        S[(size_t)(m0 + v + 8 * h) * D + n0 + ml] = c[v];
    }
}

// out[n][d] = b[d]  (bias folded into the accumulator init)
__global__ void __launch_bounds__(256)
init_out(float* __restrict__ out, const float* __restrict__ b, int total)
{
    int i = blockIdx.x * 256 + threadIdx.x;
    if (i < total) out[i] = b[i & (D - 1)];
}

// For each edge e: out[row[e], :] += edge_vals[e] * support[col[e], :]
// 2 edges per wave; 16 lanes per edge; 1 float4 gather + 4 f32 hw atomics per lane.
__global__ void __launch_bounds__(256)
edge_scatter(const float* __restrict__ S, const float* __restrict__ ev,
             const int* __restrict__ row, const int* __restrict__ col,
             float* __restrict__ out, int E)
{
    const int tid  = blockIdx.x * 256 + threadIdx.x;
    const int wid  = tid >> 5;
    const int lane = threadIdx.x & 31;
    const int h    = lane >> 4;                   // which edge within the wave
    const int li   = lane & 15;                   // 16 lanes cover 64 floats as float4

    const int e = wid * 2 + h;
    if (e >= E) return;

    const int   r = row[e];
    const int   c = col[e];
    const float v = ev[e];

    const float4 s = *(const float4*)(S + (size_t)c * D + li * 4);
    float* o = out + (size_t)r * D + li * 4;
    unsafeAtomicAdd(o + 0, v * s.x);
    unsafeAtomicAdd(o + 1, v * s.y);
    unsafeAtomicAdd(o + 2, v * s.z);
    unsafeAtomicAdd(o + 3, v * s.w);
}

extern "C" void kernel_launch(void* const* d_in, const int* in_sizes, int n_in,
                              void* d_out, int out_size, void* d_ws, size_t ws_size,
                              hipStream_t stream)
{
    const float* X   = (const float*)d_in[0];   // (N, 64)
    const float* W   = (const float*)d_in[1];   // (64, 64)
    const float* b   = (const float*)d_in[2];   // (64,)
    const float* ev  = (const float*)d_in[3];   // (E,)
    const int*   row = (const int*)d_in[4];     // (E,)
    const int*   col = (const int*)d_in[5];     // (E,)
    float*       out = (float*)d_out;           // (N, 64)
    float*       S   = (float*)d_ws;            // scratch: support (N, 64) = 25.6 MB

    const int N = in_sizes[0] / D;
    const int E = in_sizes[3];

    // 1) support = X @ W  (WMMA f32)
    const int rowTiles  = (N + 15) / 16;
    const int gemmBlks  = (rowTiles + 1) / 2;
    gemm_wmma<<<gemmBlks, 256, 0, stream>>>(X, W, S, N);

    // 2) out = broadcast(b)
    const int total    = N * D;
    const int initBlks = (total + 255) / 256;
    init_out<<<initBlks, 256, 0, stream>>>(out, b, total);

    // 3) scatter-add over edges
    const int waves    = (E + 1) / 2;
    const int edgeBlks = (waves + 7) / 8;       // 8 waves per 256-thread block
    edge_scatter<<<edgeBlks, 256, 0, stream>>>(S, ev, row, col, out, E);
}